// APMLSparse_51874615001116
// MI455X (gfx1250) — compile-verified
//
#include <hip/hip_runtime.h>
#include <math.h>

typedef float v2f __attribute__((ext_vector_type(2)));
typedef float v4f __attribute__((ext_vector_type(4)));
typedef float v8f __attribute__((ext_vector_type(8)));

#define B_     4
#define N_     4096
#define M_     4096
#define TN     16            // n-rows per block (one WMMA tile of columns)
#define PITCH  4100          // multiple of 4 (16B-aligned rows), 4100%64=4 staggers banks
#define NWAVE  16
#define BLK    (NWAVE * 32)
#define PMINF  0.8f
#define THRESH 1e-10f
#define BS_ITERS 40

__device__ __forceinline__ float wave_sum(float v) {
#pragma unroll
    for (int off = 16; off >= 1; off >>= 1) v += __shfl_xor(v, off, 32);
    return v;
}
__device__ __forceinline__ float wave_min(float v) {
#pragma unroll
    for (int off = 16; off >= 1; off >>= 1) v = fminf(v, __shfl_xor(v, off, 32));
    return v;
}

// One block: 16 consecutive n-rows of one batch. Phase 1: WMMA Gram -> d into LDS.
// Phase 2-4: each wave privately owns one row: softmax, threshold binary search, loss.
__global__ void __launch_bounds__(BLK)
apml_row_kernel(const float* __restrict__ x, const float* __restrict__ y,
                float* __restrict__ partial)
{
    extern __shared__ float smem[];
    float* e_ls  = smem;                 // [TN][PITCH]  distances, then e = exp(dmin-d)
    float* yy_ls = smem + TN * PITCH;    // [M_]         ||y_m||^2
    __shared__ float red_min[NWAVE][32];
    __shared__ float rowloss[TN];

    const int tid   = threadIdx.x;
    const int lane  = tid & 31;
    const int w     = tid >> 5;
    const int rowsB = N_ / TN;                       // 256 blocks per batch
    const int batch = blockIdx.x / rowsB;
    const int n0    = (blockIdx.x % rowsB) * TN;

    const float* xb = x + (size_t)batch * N_ * 3;
    const float* yb = y + (size_t)batch * M_ * 3;

    // ---- precompute ||y_m||^2 into LDS (coalesced) ----
    for (int m = tid; m < M_; m += BLK) {
        float y0 = yb[m * 3 + 0], y1 = yb[m * 3 + 1], y2 = yb[m * 3 + 2];
        yy_ls[m] = y0 * y0 + y1 * y1 + y2 * y2;
    }

    // ---- per-lane x column (B fragment, K padded 3->4 with zero) ----
    const int  jcol = lane & 15;
    const bool half = (lane >= 16);
    const float x0 = xb[(n0 + jcol) * 3 + 0];
    const float x1 = xb[(n0 + jcol) * 3 + 1];
    const float x2 = xb[(n0 + jcol) * 3 + 2];
    const float xx = x0 * x0 + x1 * x1 + x2 * x2;
    v2f bfrag;                       // lanes 0-15: K=0,1 ; lanes 16-31: K=2,3(=0)
    bfrag.x = half ? x2 : x0;
    bfrag.y = half ? 0.0f : x1;

    __syncthreads();

    // ---- phase 1: distances via V_WMMA_F32_16X16X4_F32 ----
    float dmin_l = 3.4e38f;
    for (int mt = w; mt < M_ / 16; mt += NWAVE) {     // uniform 16 iters per wave
        const int m0   = mt * 16;
        const int mrow = m0 + jcol;                   // A row = lane%16 (both halves)
        v2f afrag;                                    // lanes 0-15: K=0,1 ; 16-31: K=2,3
        afrag.x = half ? yb[mrow * 3 + 2] : yb[mrow * 3 + 0];
        afrag.y = half ? 0.0f             : yb[mrow * 3 + 1];
        v8f acc = {};
        acc = __builtin_amdgcn_wmma_f32_16x16x4_f32(
            false, afrag, false, bfrag, (short)0, acc, false, false);

        // lane's 8 results are m = mbase..mbase+7 (consecutive) in column jcol
        const int mbase = m0 + (half ? 8 : 0);
        v4f yyA = *(const v4f*)(yy_ls + mbase);
        v4f yyB = *(const v4f*)(yy_ls + mbase + 4);
        v4f dA, dB;
#pragma unroll
        for (int k = 0; k < 4; ++k) {
            float sqA = xx + yyA[k] - 2.0f * acc[k];
            float sqB = xx + yyB[k] - 2.0f * acc[k + 4];
            dA[k] = __builtin_amdgcn_sqrtf(fmaxf(sqA, 1e-12f));   // EPS^2 clamp, hw sqrt
            dB[k] = __builtin_amdgcn_sqrtf(fmaxf(sqB, 1e-12f));
            dmin_l = fminf(dmin_l, fminf(dA[k], dB[k]));
        }
        float* dst = e_ls + jcol * PITCH + mbase;     // 16B aligned (PITCH%4==0, mbase%8==0)
        *(v4f*)(dst)     = dA;
        *(v4f*)(dst + 4) = dB;
    }
    red_min[w][lane] = dmin_l;
    __syncthreads();

    // ---- phases 2-4: wave w owns row j = w ----
    const int j = w;
    float* row = e_ls + (size_t)j * PITCH;

    // row min over the 32 partials belonging to column j
    float dmin = wave_min(red_min[lane & 15][j + (half ? 16 : 0)]);

    // softmax numerators e = exp(dmin - d) (max e == 1), partition sum Z  (b128 LDS)
    float zp = 0.0f;
    for (int i = lane * 4; i < M_; i += 128) {
        v4f dd = *(const v4f*)(row + i);
        v4f e;
#pragma unroll
        for (int k = 0; k < 4; ++k) { e[k] = __expf(dmin - dd[k]); zp += e[k]; }
        *(v4f*)(row + i) = e;
    }
    const float Z = wave_sum(zp);

    // binary search threshold: keep iff S(e_j) = sum_{e_k > e_j} e_k < 0.8*Z
    float lo = 0.0f, hiT = 1.0f;                      // S(0)=Z>=t, S(1)=0<t
    const float target = PMINF * Z;
    for (int it = 0; it < BS_ITERS; ++it) {
        float mid = 0.5f * (lo + hiT);
        float sp = 0.0f;
        for (int i = lane * 4; i < M_; i += 128) {
            v4f e = *(const v4f*)(row + i);
#pragma unroll
            for (int k = 0; k < 4; ++k) sp += (e[k] > mid) ? e[k] : 0.0f;
        }
        float S = wave_sum(sp);
        if (S >= target) lo = mid; else hiT = mid;
    }

    // loss: sum of p*d over kept entries; d = dmin - log(e)
    const float tkeep = fmaxf(lo, THRESH * Z);
    float lp = 0.0f;
    for (int i = lane * 4; i < M_; i += 128) {
        v4f e = *(const v4f*)(row + i);
#pragma unroll
        for (int k = 0; k < 4; ++k)
            if (e[k] > tkeep) lp += e[k] * (dmin - __logf(e[k]));
    }
    float Lj = wave_sum(lp);
    if (lane == 0) rowloss[j] = Lj / Z;
    __syncthreads();

    if (tid == 0) {
        float s = 0.0f;
#pragma unroll
        for (int jj = 0; jj < TN; ++jj) s += rowloss[jj];
        partial[blockIdx.x] = s;
    }
}

// deterministic fixed-order final reduction of the 1024 block partials
__global__ void __launch_bounds__(256)
apml_final_reduce(const float* __restrict__ partial, float* __restrict__ out, int n)
{
    __shared__ float s[256];
    int t = threadIdx.x;
    float v = 0.0f;
    for (int i = t; i < n; i += 256) v += partial[i];
    s[t] = v;
    __syncthreads();
    for (int off = 128; off > 0; off >>= 1) {
        if (t < off) s[t] += s[t + off];
        __syncthreads();
    }
    if (t == 0) out[0] = s[0];
}

extern "C" void kernel_launch(void* const* d_in, const int* in_sizes, int n_in,
                              void* d_out, int out_size, void* d_ws, size_t ws_size,
                              hipStream_t stream) {
    (void)in_sizes; (void)n_in; (void)out_size; (void)ws_size;
    const float* x = (const float*)d_in[0];
    const float* y = (const float*)d_in[1];
    float* partial = (float*)d_ws;

    const int    nblocks = (B_ * N_) / TN;                            // 1024
    const size_t shbytes = (size_t)(TN * PITCH + M_) * sizeof(float); // ~272 KB

    // CDNA5 allows up to 320 KB LDS per workgroup; raise the dynamic-LDS cap.
    hipFuncSetAttribute((const void*)apml_row_kernel,
                        hipFuncAttributeMaxDynamicSharedMemorySize, (int)shbytes);

    apml_row_kernel<<<nblocks, BLK, shbytes, stream>>>(x, y, partial);
    apml_final_reduce<<<1, 256, 0, stream>>>(partial, (float*)d_out, nblocks);
}